// Sparsemax_89043261981286
// MI455X (gfx1250) — compile-verified
//
#include <hip/hip_runtime.h>
#include <stdint.h>

// Sparsemax over rows of N=4096 f32, one workgroup (256 threads, 8 wave32) per row.
// Row is staged global->LDS with CDNA5 async copies, then held in registers while
// Michelot's fixed-point iteration finds the simplex threshold tau (no sort).

#define N            4096
#define THREADS      256
#define CHUNKS       4            // 4 x float4 = 16 elements per thread
#define NWAVES       (THREADS / 32)

__global__ __launch_bounds__(THREADS)
void sparsemax_rows_kernel(const float* __restrict__ x, float* __restrict__ out)
{
    __shared__ __align__(16) float srow[N];
    __shared__ float red_s[NWAVES];
    __shared__ float red_c[NWAVES];

    const int t = threadIdx.x;
    const long long rowBase = (long long)blockIdx.x * (long long)N;
    const float* __restrict__ rowIn  = x   + rowBase;
    float*       __restrict__ rowOut = out + rowBase;

    // ---------------- Stage row into LDS via gfx1250 async global->LDS ----------------
    // Each instruction: 32 lanes x 16B lane-consecutive => fully coalesced.
#pragma unroll
    for (int j = 0; j < CHUNKS; ++j) {
        const int idx = j * (THREADS * 4) + t * 4;
        uint64_t gaddr = (uint64_t)(uintptr_t)(rowIn + idx);
        uint32_t laddr = (uint32_t)(uintptr_t)(&srow[idx]);   // low 32b of generic LDS ptr = LDS byte offset
        asm volatile("global_load_async_to_lds_b128 %0, %1, off"
                     :: "v"(laddr), "v"(gaddr)
                     : "memory");
    }
    // ASYNCcnt==0 => our own LDS writes have landed (threads only read back their own bytes).
    asm volatile("s_wait_asynccnt 0" ::: "memory");

    // Pull this thread's 16 elements into registers (ds_load_b128 x4, bank-conflict free).
    float4 v[CHUNKS];
#pragma unroll
    for (int j = 0; j < CHUNKS; ++j) {
        const int idx = j * (THREADS * 4) + t * 4;
        v[j] = *reinterpret_cast<const float4*>(&srow[idx]);
    }

    // ---------------- Initial tau: tau0 = (sum(x) - 1) / N ----------------
    float s = 0.0f;
#pragma unroll
    for (int j = 0; j < CHUNKS; ++j)
        s += v[j].x + v[j].y + v[j].z + v[j].w;

#pragma unroll
    for (int m = 16; m >= 1; m >>= 1)
        s += __shfl_xor(s, m, 32);
    if ((t & 31) == 0) red_s[t >> 5] = s;
    __syncthreads();
    float S0 = 0.0f;
#pragma unroll
    for (int i = 0; i < NWAVES; ++i) S0 += red_s[i];

    float tau   = (S0 - 1.0f) * (1.0f / (float)N);
    float kprev = (float)N;

    // ---------------- Michelot fixed-point iteration ----------------
    // A = {x > tau};  tau <- (sum(A) - 1)/|A|;  stop when |A| is unchanged.
    for (int it = 0; it < 64; ++it) {
        float ps = 0.0f, pc = 0.0f;
#pragma unroll
        for (int j = 0; j < CHUNKS; ++j) {
            float e;
            e = v[j].x; if (e > tau) { ps += e; pc += 1.0f; }
            e = v[j].y; if (e > tau) { ps += e; pc += 1.0f; }
            e = v[j].z; if (e > tau) { ps += e; pc += 1.0f; }
            e = v[j].w; if (e > tau) { ps += e; pc += 1.0f; }
        }
#pragma unroll
        for (int m = 16; m >= 1; m >>= 1) {
            ps += __shfl_xor(ps, m, 32);
            pc += __shfl_xor(pc, m, 32);
        }
        __syncthreads();                       // protect red_* from previous iteration's readers
        if ((t & 31) == 0) { red_s[t >> 5] = ps; red_c[t >> 5] = pc; }
        __syncthreads();
        float S = 0.0f, C = 0.0f;
#pragma unroll
        for (int i = 0; i < NWAVES; ++i) { S += red_s[i]; C += red_c[i]; }

        if (C == kprev || C == 0.0f) break;    // active set stable -> converged (uniform branch)
        kprev = C;
        tau = (S - 1.0f) / C;
    }

    // ---------------- Epilogue: out = max(x - tau, 0), coalesced b128 stores ----------------
#pragma unroll
    for (int j = 0; j < CHUNKS; ++j) {
        const int idx = j * (THREADS * 4) + t * 4;
        float4 r;
        r.x = fmaxf(v[j].x - tau, 0.0f);
        r.y = fmaxf(v[j].y - tau, 0.0f);
        r.z = fmaxf(v[j].z - tau, 0.0f);
        r.w = fmaxf(v[j].w - tau, 0.0f);
        *reinterpret_cast<float4*>(rowOut + idx) = r;
    }
}

extern "C" void kernel_launch(void* const* d_in, const int* in_sizes, int n_in,
                              void* d_out, int out_size, void* d_ws, size_t ws_size,
                              hipStream_t stream)
{
    const float* x   = (const float*)d_in[0];
    float*       out = (float*)d_out;
    const int rows   = in_sizes[0] / N;       // 16384 for the reference shapes
    sparsemax_rows_kernel<<<dim3(rows), dim3(THREADS), 0, stream>>>(x, out);
}